// HallucinatorLoss_86629490360588
// MI455X (gfx1250) — compile-verified
//
#include <hip/hip_runtime.h>
#include <cstdint>

// ---------------------------------------------------------------------------
// HallucinatorLoss: sum over rows of (1 - sum(top8(row))) for a 16384x8192 f32
// matrix. Memory-bound streaming reduce (512 MB @ 23.3 TB/s ~ 22 us floor).
// CDNA5 paths: async global->LDS DMA (ASYNCcnt) double buffering, ds_load_b128
// tile drains, wave32 shuffle/ballot merge. Top-8 maintenance uses raw
// v_med3_num_f32 (inline asm, avoids NaN-legalization expansion): 8 VALU
// ops/element, ~1.07 G lane-ops total — below the HBM roofline.
// ---------------------------------------------------------------------------

#define COLS            8192
#define K               8
#define TILE            512                 // floats per tile per wave (2 KB)
#define J               (TILE / 4 / 32)     // float4 async ops per lane = 4
#define NT              (COLS / TILE)       // tiles per row = 16
#define WAVES_PER_BLOCK 8
#define BLOCK_THREADS   (WAVES_PER_BLOCK * 32)

#define NEG_HUGE -3.0e38f

// ---- CDNA5 async global->LDS copy (per-lane 16B), portable via inline asm ---
__device__ __forceinline__ void async_ld_b128(uint32_t lds_byte, const float* g) {
    asm volatile("global_load_async_to_lds_b128 %0, %1, off"
                 :
                 : "v"(lds_byte), "v"((unsigned long long)(uintptr_t)g)
                 : "memory");
}
#define WAIT_ASYNC_LE(N) asm volatile("s_wait_asynccnt " #N ::: "memory")

// Single-instruction 3-input median (no NaN-quieting expansion).
__device__ __forceinline__ float med3(float x, float a, float b) {
    float d;
    asm("v_med3_num_f32 %0, %1, %2, %3" : "=v"(d) : "v"(x), "v"(a), "v"(b));
    return d;
}

// Insert x into a descending-sorted 8-register top list using med3:
//   t[i]' = med3(x, t[i-1], t[i])  (t[i-1] >= t[i] clamps x into its rank slot)
//   t[0]' = max(t[0], x)
// Updating i = 7..1 consumes the pre-update t[i-1]. 8 full-rate VALU ops,
// mutually independent within one insertion (good dual-issue fodder).
__device__ __forceinline__ void insert8(float (&tk)[K], float x) {
#pragma unroll
    for (int i = K - 1; i >= 1; --i)
        tk[i] = med3(x, tk[i - 1], tk[i]);
    tk[0] = fmaxf(tk[0], x);
}

__global__ void __launch_bounds__(BLOCK_THREADS)
hallucinator_loss_init(float* out, int rows) {
    if (threadIdx.x == 0 && blockIdx.x == 0) out[0] = (float)rows;
}

__global__ void __launch_bounds__(BLOCK_THREADS)
hallucinator_loss_kernel(const float* __restrict__ vals, float* __restrict__ out) {
    __shared__ __align__(16) float ldsBuf[WAVES_PER_BLOCK * 2 * TILE];
    __shared__ float sPartial[WAVES_PER_BLOCK];

    const int lane = threadIdx.x & 31;
    const int wave = threadIdx.x >> 5;
    const int row  = blockIdx.x * WAVES_PER_BLOCK + wave;

    const float* rowPtr = vals + (size_t)row * COLS;

    float* buf0 = &ldsBuf[wave * 2 * TILE];
    float* buf1 = buf0 + TILE;
    // Low 32 bits of a generic LDS pointer == LDS byte offset (ISA 10.2).
    const uint32_t base0 = (uint32_t)(uintptr_t)buf0;
    const uint32_t base1 = base0 + TILE * 4u;

    // Per-lane running top-8 (descending; tk[7] is the current minimum).
    float tk[K];
#pragma unroll
    for (int i = 0; i < K; ++i) tk[i] = NEG_HUGE;

    // ---- issue first two tiles (async DMA, no VGPR data path) ----
    {
        const float* s0 = rowPtr + 4u * lane;
        const float* s1 = s0 + TILE;
        const uint32_t d0 = base0 + 16u * lane;
        const uint32_t d1 = base1 + 16u * lane;
#pragma unroll
        for (int j = 0; j < J; ++j) {
            async_ld_b128(d0 + 512u * j, s0 + 128 * j);
        }
#pragma unroll
        for (int j = 0; j < J; ++j) {
            async_ld_b128(d1 + 512u * j, s1 + 128 * j);
        }
    }

    for (int t = 0; t < NT; ++t) {
        // Async loads retire in order: with two tiles (2*J ops) in flight,
        // waiting to <=J guarantees the current tile landed in LDS.
        if (t + 1 < NT) { WAIT_ASYNC_LE(4); }
        else            { WAIT_ASYNC_LE(0); }

        const float*   cur     = (t & 1) ? buf1 : buf0;
        const uint32_t curBase = (t & 1) ? base1 : base0;

#pragma unroll
        for (int j = 0; j < J; ++j) {
            const float4 v = ((const float4*)cur)[j * 32 + lane];
            insert8(tk, v.x);
            insert8(tk, v.y);
            insert8(tk, v.z);
            insert8(tk, v.w);
        }
        // All ds reads above were consumed before we overwrite the buffer.
        if (t + 2 < NT) {
            const float* s = rowPtr + (size_t)(t + 2) * TILE + 4u * lane;
            const uint32_t d = curBase + 16u * lane;
#pragma unroll
            for (int j = 0; j < J; ++j) {
                async_ld_b128(d + 512u * j, s + 128 * j);
            }
        }
    }

    // ---- wave32 merge: extract the row's global top-8 from 32 sorted lists --
    float rowSum = 0.0f;
#pragma unroll
    for (int k = 0; k < K; ++k) {
        float m = tk[0];
#pragma unroll
        for (int off = 16; off > 0; off >>= 1)
            m = fmaxf(m, __shfl_xor(m, off, 32));
        rowSum += m;
        unsigned long long ball = __ballot(tk[0] == m);
        int src = (int)__ffsll(ball) - 1;
        if (lane == src) {
#pragma unroll
            for (int i = 0; i < K - 1; ++i) tk[i] = tk[i + 1];
            tk[K - 1] = NEG_HUGE;
        }
    }

    if (lane == 0) sPartial[wave] = rowSum;
    __syncthreads();

    if (threadIdx.x == 0) {
        float acc = 0.0f;
#pragma unroll
        for (int w = 0; w < WAVES_PER_BLOCK; ++w) acc += sPartial[w];
        // out was seeded with (float)rows; loss = rows - sum(rowsums).
        atomicAdd(out, -acc);
    }
}

extern "C" void kernel_launch(void* const* d_in, const int* in_sizes, int n_in,
                              void* d_out, int out_size, void* d_ws, size_t ws_size,
                              hipStream_t stream) {
    const float* vals = (const float*)d_in[0];
    float* out = (float*)d_out;

    const int rows = in_sizes[0] / COLS;           // 16384
    const int blocks = rows / WAVES_PER_BLOCK;     // 2048

    hallucinator_loss_init<<<1, BLOCK_THREADS, 0, stream>>>(out, rows);
    hallucinator_loss_kernel<<<blocks, BLOCK_THREADS, 0, stream>>>(vals, out);
}